// EmbeddingBagCollection_34291018891792
// MI455X (gfx1250) — compile-verified
//
#include <hip/hip_runtime.h>

// EmbeddingBagCollection (4 tables, mean pooling) for MI455X / gfx1250.
//
// Bandwidth-bound gather: ~435 MB moved -> ~19 us at 23.3 TB/s HBM.
// Strategy (wave32-native):
//   * grid = (1024, 4): blockIdx.y selects the table (wave-uniform -> pure
//     SALU branch, pointers live in SGPRs), 8 waves/block, one wave per bag
//   * half-wave (16 lanes) gathers one 256B embedding row via float4
//     (global_load_b128) -> one wave fetches TWO rows per vmem instruction
//   * 8 rows in flight per unrolled iteration to maximize outstanding loads
//   * global_prefetch_b8 on next iteration's row pair (gfx1250 prefetch path)
//   * W gathers use default (RT) temporal hint: ~20% repeat rows + 192MB L2
//     reward caching; output stores are nontemporal (written once)
//   * half-wave partials combined with __shfl_xor(.,16); lanes 0-15 emit one
//     coalesced 256B row store

#define BAGS 8192
#define DIM 64
#define NT 4
#define WAVES_PER_BLOCK 8
#define THREADS (WAVES_PER_BLOCK * 32)

__global__ __launch_bounds__(THREADS) void ebc_pool_mean_kernel(
    const float* __restrict__ W0, const int* __restrict__ V0, const int* __restrict__ O0,
    const float* __restrict__ W1, const int* __restrict__ V1, const int* __restrict__ O1,
    const float* __restrict__ W2, const int* __restrict__ V2, const int* __restrict__ O2,
    const float* __restrict__ W3, const int* __restrict__ V3, const int* __restrict__ O3,
    float* __restrict__ out)
{
    const int lane = threadIdx.x & 31;
    const int half = lane >> 4;    // 0: even rows of the bag, 1: odd rows
    const int sub  = lane & 15;    // which float4 chunk of the 64-wide row
    const int bag  = blockIdx.x * WAVES_PER_BLOCK + (threadIdx.x >> 5);
    const int table = blockIdx.y;  // wave-uniform -> scalar branch below

    const float* W;
    const int*   V;
    const int*   O;
    switch (table) {
        case 0:  W = W0; V = V0; O = O0; break;
        case 1:  W = W1; V = V1; O = O1; break;
        case 2:  W = W2; V = V2; O = O2; break;
        default: W = W3; V = V3; O = O3; break;
    }

    const int start = O[bag];
    const int end   = O[bag + 1];
    const int count = end - start;
    const int col   = sub << 2;            // float offset within row (16B aligned)

    float4 acc = make_float4(0.f, 0.f, 0.f, 0.f);

    int i = start;
    // Main loop: 8 rows per iteration; each half-wave owns 4 of them as b128 gathers.
    for (; i + 7 < end; i += 8) {
        const int idx0 = V[i + 0 + half];
        const int idx1 = V[i + 2 + half];
        const int idx2 = V[i + 4 + half];
        const int idx3 = V[i + 6 + half];

        // Pull next iteration's first row pair toward GL2 early.
        if (i + 8 + half < end) {
            const int pidx = V[i + 8 + half];
            __builtin_prefetch(W + (size_t)pidx * DIM + col, 0, 1);
        }

        const float4 r0 = *(const float4*)(W + (size_t)idx0 * DIM + col);
        const float4 r1 = *(const float4*)(W + (size_t)idx1 * DIM + col);
        const float4 r2 = *(const float4*)(W + (size_t)idx2 * DIM + col);
        const float4 r3 = *(const float4*)(W + (size_t)idx3 * DIM + col);

        acc.x += r0.x; acc.y += r0.y; acc.z += r0.z; acc.w += r0.w;
        acc.x += r1.x; acc.y += r1.y; acc.z += r1.z; acc.w += r1.w;
        acc.x += r2.x; acc.y += r2.y; acc.z += r2.z; acc.w += r2.w;
        acc.x += r3.x; acc.y += r3.y; acc.z += r3.z; acc.w += r3.w;
    }
    // Remaining full pairs.
    for (; i + 1 < end; i += 2) {
        const int idx = V[i + half];
        const float4 r = *(const float4*)(W + (size_t)idx * DIM + col);
        acc.x += r.x; acc.y += r.y; acc.z += r.z; acc.w += r.w;
    }
    // Odd trailing row: only the even half-wave accumulates it (avoid double count).
    if (i < end && half == 0) {
        const int idx = V[i];
        const float4 r = *(const float4*)(W + (size_t)idx * DIM + col);
        acc.x += r.x; acc.y += r.y; acc.z += r.z; acc.w += r.w;
    }

    // Combine even-row and odd-row partial sums across the two half-waves (wave32).
    acc.x += __shfl_xor(acc.x, 16, 32);
    acc.y += __shfl_xor(acc.y, 16, 32);
    acc.z += __shfl_xor(acc.z, 16, 32);
    acc.w += __shfl_xor(acc.w, 16, 32);

    if (half == 0) {
        const float denom = (float)(count > 0 ? count : 1);
        // KeyedTensor layout: out[bag, table*64 + dim], row stride 256 floats.
        float* dst = out + (size_t)bag * (NT * DIM) + table * DIM + col;
        __builtin_nontemporal_store(acc.x / denom, dst + 0);
        __builtin_nontemporal_store(acc.y / denom, dst + 1);
        __builtin_nontemporal_store(acc.z / denom, dst + 2);
        __builtin_nontemporal_store(acc.w / denom, dst + 3);
    }
}

extern "C" void kernel_launch(void* const* d_in, const int* in_sizes, int n_in,
                              void* d_out, int out_size, void* d_ws, size_t ws_size,
                              hipStream_t stream)
{
    (void)in_sizes; (void)n_in; (void)out_size; (void)d_ws; (void)ws_size;

    // setup_inputs() dict order: W0, values0, offsets0, W1, values1, offsets1, ...
    const float* W0 = (const float*)d_in[0];
    const int*   V0 = (const int*)  d_in[1];
    const int*   O0 = (const int*)  d_in[2];
    const float* W1 = (const float*)d_in[3];
    const int*   V1 = (const int*)  d_in[4];
    const int*   O1 = (const int*)  d_in[5];
    const float* W2 = (const float*)d_in[6];
    const int*   V2 = (const int*)  d_in[7];
    const int*   O2 = (const int*)  d_in[8];
    const float* W3 = (const float*)d_in[9];
    const int*   V3 = (const int*)  d_in[10];
    const int*   O3 = (const int*)  d_in[11];

    dim3 grid(BAGS / WAVES_PER_BLOCK, NT);   // (1024, 4)

    ebc_pool_mean_kernel<<<grid, THREADS, 0, stream>>>(
        W0, V0, O0, W1, V1, O1, W2, V2, O2, W3, V3, O3, (float*)d_out);
}